// HierGraphAttNet_35021163332296
// MI455X (gfx1250) — compile-verified
//
#include <hip/hip_runtime.h>
#include <math.h>

// Problem constants (B, S, D) = (4096, 10, 1024)
#define B_N   4096
#define S_N   10
#define D_N   1024
#define K_BIG 2048   // 2*D
#define N1    1024   // Wg output dim
#define N2    512    // Wfd output dim

typedef __attribute__((ext_vector_type(2))) float v2f;
typedef __attribute__((ext_vector_type(8))) float v8f;

// Workspace layout (float offsets)
#define R_OFF     0                          // r scores: B*16 (padded 10->16)
#define MAX_OFF   (B_N * 16)                 // 16 per-t maxima
#define SUM_OFF   (MAX_OFF + 16)             // 16 per-t exp-sums
#define LOGIT_OFF (SUM_OFF + 16)             // B logits
#define AX_OFF    (LOGIT_OFF + B_N)          // attn_x last-row: B*D
#define AY_OFF    (AX_OFF + (size_t)B_N*D_N) // attn_y last-row: B*D
#define O_OFF     (AY_OFF + (size_t)B_N*D_N) // O = [o1 rows; o2 rows]: 2B*N1

// ---------------------------------------------------------------------------
// Kernel 1: per-batch scores (both directions) + local softmax + attn_x.
// One block per batch; streams X1[b], X2[b] exactly once from HBM.
// ---------------------------------------------------------------------------
__global__ __launch_bounds__(256) void attn_prep(const float* __restrict__ x1,
                                                 const float* __restrict__ x2,
                                                 float* __restrict__ ws) {
  const int b   = blockIdx.x;
  const int tid = threadIdx.x;
  const float* X1 = x1 + (size_t)b * S_N * D_N;
  const float* X2 = x2 + (size_t)b * S_N * D_N;

  __shared__ float lds_x1[S_N * D_N];   // 40 KB: cache X1[b] for the weighted sum
  __shared__ float red[20 * 8];
  __shared__ float sc[20];
  __shared__ float wgt[S_N];

  float cp[S_N], rp[S_N];
#pragma unroll
  for (int s = 0; s < S_N; ++s) { cp[s] = 0.f; rp[s] = 0.f; }

  for (int d = tid; d < D_N; d += 256) {
    const float v1l = X1[(S_N - 1) * D_N + d];
    const float v2l = X2[(S_N - 1) * D_N + d];
#pragma unroll
    for (int s = 0; s < S_N; ++s) {
      const float a = X1[s * D_N + d];
      lds_x1[s * D_N + d] = a;
      cp[s] += a * v2l;                 // c[s] = <X1[b,s], X2[b,S-1]>
      rp[s] += X2[s * D_N + d] * v1l;   // r[t] = <X1[b,S-1], X2[b,t]>
    }
  }

  const int lane = tid & 31, wid = tid >> 5;   // wave32 on gfx1250
#pragma unroll
  for (int v = 0; v < S_N; ++v) {
    float c = cp[v], r = rp[v];
#pragma unroll
    for (int off = 16; off > 0; off >>= 1) {
      c += __shfl_down(c, off, 32);
      r += __shfl_down(r, off, 32);
    }
    if (lane == 0) { red[v * 8 + wid] = c; red[(S_N + v) * 8 + wid] = r; }
  }
  __syncthreads();

  if (tid < 20) {
    float s = 0.f;
#pragma unroll
    for (int w = 0; w < 8; ++w) s += red[tid * 8 + w];
    sc[tid] = s;
    if (tid >= S_N) ws[R_OFF + (size_t)b * 16 + (tid - S_N)] = s;  // stash r[t]
  }
  __syncthreads();

  if (tid == 0) {  // softmax over s for the a_x column t=S-1
    float m = sc[0];
    for (int s = 1; s < S_N; ++s) m = fmaxf(m, sc[s]);
    float e[S_N], sum = 0.f;
    for (int s = 0; s < S_N; ++s) { e[s] = expf(sc[s] - m); sum += e[s]; }
    const float inv = 1.f / sum;
    for (int s = 0; s < S_N; ++s) wgt[s] = e[s] * inv;
  }
  __syncthreads();

  float* ax = ws + AX_OFF + (size_t)b * D_N;
  for (int d = tid; d < D_N; d += 256) {
    float acc = 0.f;
#pragma unroll
    for (int s = 0; s < S_N; ++s) acc += wgt[s] * lds_x1[s * D_N + d];
    ax[d] = acc;
  }
}

// ---------------------------------------------------------------------------
// Kernel 2: batch-axis softmax statistics (max + sum of exp over B) per t.
// ---------------------------------------------------------------------------
__global__ __launch_bounds__(256) void batch_stats(float* __restrict__ ws) {
  const int t   = blockIdx.x;      // 0..S_N-1
  const int tid = threadIdx.x;
  __shared__ float red[256];

  float m = -3.402823466e38f;
  for (int b = tid; b < B_N; b += 256) m = fmaxf(m, ws[R_OFF + (size_t)b * 16 + t]);
  red[tid] = m; __syncthreads();
  for (int off = 128; off > 0; off >>= 1) {
    if (tid < off) red[tid] = fmaxf(red[tid], red[tid + off]);
    __syncthreads();
  }
  const float gm = red[0];
  __syncthreads();

  float s = 0.f;
  for (int b = tid; b < B_N; b += 256) s += expf(ws[R_OFF + (size_t)b * 16 + t] - gm);
  red[tid] = s; __syncthreads();
  for (int off = 128; off > 0; off >>= 1) {
    if (tid < off) red[tid] += red[tid + off];
    __syncthreads();
  }
  if (tid == 0) { ws[MAX_OFF + t] = gm; ws[SUM_OFF + t] = red[0]; }
}

// ---------------------------------------------------------------------------
// Kernel 3: attn_y last-row = sum_t softmax_batch(r)[b,t] * X2[b,t,:]
// ---------------------------------------------------------------------------
__global__ __launch_bounds__(256) void attn_y_kernel(const float* __restrict__ x2,
                                                     float* __restrict__ ws) {
  const int b   = blockIdx.x;
  const int tid = threadIdx.x;
  __shared__ float wgt[S_N];
  if (tid < S_N)
    wgt[tid] = expf(ws[R_OFF + (size_t)b * 16 + tid] - ws[MAX_OFF + tid]) / ws[SUM_OFF + tid];
  __syncthreads();

  const float* X2 = x2 + (size_t)b * S_N * D_N;
  float* ay = ws + AY_OFF + (size_t)b * D_N;
  for (int d = tid; d < D_N; d += 256) {
    float acc = 0.f;
#pragma unroll
    for (int t = 0; t < S_N; ++t) acc += wgt[t] * X2[t * D_N + d];
    ay[d] = acc;
  }
}

// ---------------------------------------------------------------------------
// Kernel 4: WMMA f32 GEMM:  O = A @ Wg + bg,  M=2B, N=1024, K=2048.
// Virtual A: rows [0,B)   = [X1[b,S-1,:] | attn_y[b,:]]   -> o1 rows
//            rows [B,2B)  = [X2[b,S-1,:] | attn_x[b,:]]   -> o2 rows
// 4 waves/block, each wave owns one 16x16 N-tile; V_WMMA_F32_16X16X4_F32.
// ---------------------------------------------------------------------------
__global__ __launch_bounds__(128) void gemm1(const float* __restrict__ x1,
                                             const float* __restrict__ x2,
                                             const float* __restrict__ Wg,
                                             const float* __restrict__ bg,
                                             float* __restrict__ ws) {
  const int lane = threadIdx.x & 31;
  const int wid  = threadIdx.x >> 5;
  const int half = lane >> 4;     // 0: lanes 0-15, 1: lanes 16-31
  const int ln   = lane & 15;
  const int row0 = blockIdx.y * 16;
  const int col0 = blockIdx.x * 64 + wid * 16;
  const bool second = (row0 >= B_N);
  const int  bbase  = second ? (row0 - B_N) : row0;
  const float* xin  = second ? x2 : x1;
  const float* attn = ws + (second ? AX_OFF : AY_OFF);

  v8f acc = {};
  const int bb = bbase + ln;   // batch index for this lane's A row
  for (int k0 = 0; k0 < K_BIG; k0 += 4) {
    const int ka = k0 + 2 * half;   // A frag: VGPR0 = K=ka, VGPR1 = K=ka+1 (ISA 16x4 f32 layout)
    v2f a;
    if (ka < D_N) {
      const float* p = xin + (size_t)bb * S_N * D_N + (size_t)(S_N - 1) * D_N + ka;
      a.x = p[0]; a.y = p[1];
    } else {
      const float* p = attn + (size_t)bb * D_N + (ka - D_N);
      a.x = p[0]; a.y = p[1];
    }
    const float* q = Wg + (size_t)ka * N1 + col0 + ln;   // B frag (4x16, K rows striped)
    v2f bf; bf.x = q[0]; bf.y = q[N1];
    acc = __builtin_amdgcn_wmma_f32_16x16x4_f32(false, a, false, bf,
                                                (short)0, acc, false, false);
  }

  const int n = col0 + ln;
  const float bias = bg[n];
  float* O = ws + O_OFF;
#pragma unroll
  for (int j = 0; j < 8; ++j) {       // C/D layout: VGPR j -> M = j (+8 for upper half)
    const int m = row0 + j + (half ? 8 : 0);
    O[(size_t)m * N1 + n] = acc[j] + bias;
  }
}

// ---------------------------------------------------------------------------
// Kernel 5: WMMA f32 GEMM: h = relu([o1|o2] @ Wfd + bfd), fused logits += h*Wff.
// M=B, N=512, K=2048.  A2[b,k] = k<D ? O[b,k] : O[B+b, k-D].
// ---------------------------------------------------------------------------
__global__ __launch_bounds__(128) void gemm2(const float* __restrict__ Wfd,
                                             const float* __restrict__ bfd,
                                             const float* __restrict__ Wff,
                                             float* __restrict__ ws) {
  const int lane = threadIdx.x & 31;
  const int wid  = threadIdx.x >> 5;
  const int half = lane >> 4;
  const int ln   = lane & 15;
  const int row0 = blockIdx.y * 16;
  const int col0 = blockIdx.x * 64 + wid * 16;
  const float* O = ws + O_OFF;

  v8f acc = {};
  const int bb = row0 + ln;
  for (int k0 = 0; k0 < K_BIG; k0 += 4) {
    const int ka = k0 + 2 * half;
    v2f a;
    if (ka < D_N) { const float* p = O + (size_t)bb * N1 + ka;                     a.x = p[0]; a.y = p[1]; }
    else          { const float* p = O + (size_t)(B_N + bb) * N1 + (ka - D_N);     a.x = p[0]; a.y = p[1]; }
    const float* q = Wfd + (size_t)ka * N2 + col0 + ln;
    v2f bf; bf.x = q[0]; bf.y = q[N2];
    acc = __builtin_amdgcn_wmma_f32_16x16x4_f32(false, a, false, bf,
                                                (short)0, acc, false, false);
  }

  const int n = col0 + ln;
  const float bias = bfd[n];
  const float wf   = Wff[n];
  float* logits = ws + LOGIT_OFF;
#pragma unroll
  for (int j = 0; j < 8; ++j) {
    float h = fmaxf(acc[j] + bias, 0.f);
    float contrib = h * wf;
#pragma unroll
    for (int off = 8; off > 0; off >>= 1)     // reduce the 16 n-lanes of each half-wave
      contrib += __shfl_xor(contrib, off, 32);
    if (ln == 0) {
      const int m = row0 + j + (half ? 8 : 0);
      atomicAdd(&logits[m], contrib);
    }
  }
}

// ---------------------------------------------------------------------------
// Zero logits (graph-safe re-init each call) and final sigmoid.
// ---------------------------------------------------------------------------
__global__ void zero_logits(float* __restrict__ ws) {
  const int i = blockIdx.x * 256 + threadIdx.x;
  if (i < B_N) ws[LOGIT_OFF + i] = 0.f;
}

__global__ void finalize(const float* __restrict__ ws,
                         const float* __restrict__ bff,
                         float* __restrict__ out) {
  const int i = blockIdx.x * 256 + threadIdx.x;
  if (i < B_N) {
    const float l = ws[LOGIT_OFF + i] + bff[0];
    out[i] = 1.f / (1.f + expf(-l));
  }
}

extern "C" void kernel_launch(void* const* d_in, const int* in_sizes, int n_in,
                              void* d_out, int out_size, void* d_ws, size_t ws_size,
                              hipStream_t stream) {
  const float* x1  = (const float*)d_in[0];
  const float* x2  = (const float*)d_in[1];
  const float* Wg  = (const float*)d_in[2];
  const float* bg  = (const float*)d_in[3];
  const float* Wfd = (const float*)d_in[4];
  const float* bfd = (const float*)d_in[5];
  const float* Wff = (const float*)d_in[6];
  const float* bff = (const float*)d_in[7];
  float* out = (float*)d_out;
  float* ws  = (float*)d_ws;

  attn_prep   <<<B_N, 256, 0, stream>>>(x1, x2, ws);
  batch_stats <<<S_N, 256, 0, stream>>>(ws);
  attn_y_kernel<<<B_N, 256, 0, stream>>>(x2, ws);
  zero_logits <<<B_N / 256, 256, 0, stream>>>(ws);
  gemm1       <<<dim3(N1 / 64, (2 * B_N) / 16), 128, 0, stream>>>(x1, x2, Wg, bg, ws);
  gemm2       <<<dim3(N2 / 64, B_N / 16), 128, 0, stream>>>(Wfd, bfd, Wff, ws);
  finalize    <<<B_N / 256, 256, 0, stream>>>(ws, bff, out);
}